// SmlplingGrouping_28329604284985
// MI455X (gfx1250) — compile-verified
//
#include <hip/hip_runtime.h>
#include <hip/hip_bf16.h>

#define B_ 4
#define N_ 16384
#define S_ 2048
#define K_ 32
#define D_ 64
#define C67 (D_ + 3)   // 67 grouped channels
#define COUT (C67 + D_) // 131 output channels

typedef __attribute__((ext_vector_type(2))) float v2f;
typedef __attribute__((ext_vector_type(8))) float v8f;

// ---------------------------------------------------------------------------
// Kernel 0: zero the per-batch sum-of-squares accumulators (ws is poisoned).
// ---------------------------------------------------------------------------
__global__ void zero_acc_kernel(float* sumsq) {
    if (threadIdx.x < B_) sumsq[threadIdx.x] = 0.0f;
}

// ---------------------------------------------------------------------------
// Kernel 1: farthest point sampling. One block (1024 threads = 32 waves) per
// batch; each thread keeps 16 points' coords + running d2 in registers.
// Per iteration: min-update + argmax via wave shuffle + LDS reduction.
// ---------------------------------------------------------------------------
__global__ void __launch_bounds__(1024)
fps_kernel(const float* __restrict__ xyz, int* __restrict__ fps_idx,
           float* __restrict__ new_xyz) {
    const int b = blockIdx.x;
    const int tid = threadIdx.x;
    const int lane = tid & 31;
    const int wid = tid >> 5;
    const int PTS = N_ / 1024; // 16

    float xs[PTS], ys[PTS], zs[PTS], d2[PTS];
#pragma unroll
    for (int p = 0; p < PTS; ++p) {
        int n = tid + p * 1024;
        const float* q = xyz + ((size_t)b * N_ + n) * 3;
        xs[p] = q[0]; ys[p] = q[1]; zs[p] = q[2];
        d2[p] = 1.0e10f;
    }

    __shared__ float s_c[3];
    __shared__ int s_far;
    __shared__ float wv[32];
    __shared__ int wi[32];

    int far = 0;
    for (int it = 0; it < S_; ++it) {
        if (tid == 0) {
            fps_idx[b * S_ + it] = far;
            const float* c = xyz + ((size_t)b * N_ + far) * 3;
            float cx = c[0], cy = c[1], cz = c[2];
            s_c[0] = cx; s_c[1] = cy; s_c[2] = cz;
            float* o = new_xyz + ((size_t)b * S_ + it) * 3;
            o[0] = cx; o[1] = cy; o[2] = cz;
        }
        __syncthreads();
        float cx = s_c[0], cy = s_c[1], cz = s_c[2];

        float bestv = -1.0f; int besti = 0;
#pragma unroll
        for (int p = 0; p < PTS; ++p) {
            float dx = xs[p] - cx, dy = ys[p] - cy, dz = zs[p] - cz;
            float dist = dx * dx + dy * dy + dz * dz;
            d2[p] = fminf(d2[p], dist);
            if (d2[p] > bestv) { bestv = d2[p]; besti = tid + p * 1024; }
        }
        // wave32 reduction
#pragma unroll
        for (int off = 16; off > 0; off >>= 1) {
            float ov = __shfl_down(bestv, off);
            int   oi = __shfl_down(besti, off);
            if (ov > bestv) { bestv = ov; besti = oi; }
        }
        if (lane == 0) { wv[wid] = bestv; wi[wid] = besti; }
        __syncthreads();
        if (wid == 0) {
            bestv = wv[lane]; besti = wi[lane];
#pragma unroll
            for (int off = 16; off > 0; off >>= 1) {
                float ov = __shfl_down(bestv, off);
                int   oi = __shfl_down(besti, off);
                if (ov > bestv) { bestv = ov; besti = oi; }
            }
            if (lane == 0) s_far = besti;
        }
        __syncthreads();
        far = s_far;
    }
}

// ---------------------------------------------------------------------------
// Kernel 2: KNN via V_WMMA_F32_16X16X4_F32.
// score(m,n) = -2*q_m . x_n + ||x_n||^2   (per-row ||q||^2 dropped: rank-safe)
// A (16x4): row m = (qx,qy,qz,1);  B (4x16): col n = (-2xx,-2xy,-2xz,|x|^2)
// One wave handles a 16-query tile, 32 candidate columns per iteration
// (2 WMMAs). Selection: lane l and lane l+16 both serve row (l&15), each
// keeping a sorted top-32 sublist in LDS (dynamic shift-insert; common path
// is one register compare vs cached worst). Final 2-way merge per row.
// ---------------------------------------------------------------------------
__global__ void __launch_bounds__(256)
knn_kernel(const float* __restrict__ xyz, const float* __restrict__ new_xyz,
           int* __restrict__ knn_idx) {
    const int lane = threadIdx.x & 31;
    const int wid = threadIdx.x >> 5;
    const int tile = blockIdx.x * 8 + wid;          // 512 tiles total
    const int b = tile / (S_ / 16);
    const int s0 = (tile % (S_ / 16)) * 16;

    __shared__ float sc[8][16 * 32];       // score tiles, 2KB/wave
    __shared__ float lval[8][16 * 2 * K_]; // sorted sublist values
    __shared__ int   lidx[8][16 * 2 * K_]; // sorted sublist indices
    float* scw = sc[wid];
    float* lv = lval[wid];
    int*   li = lidx[wid];

    const int row = lane & 15;
    const int half = lane >> 4;            // which 16-col half this lane owns
    float* mylv = lv + (row * 2 + half) * K_;
    int*   myli = li + (row * 2 + half) * K_;
    for (int k = 0; k < K_; ++k) { mylv[k] = 1.0e30f; myli[k] = 0; }
    float worst = 1.0e30f;

    // Load A matrix (16x4 f32 layout: lanes 0-15 -> K0/K1, lanes 16-31 -> K2/K3)
    const float* q = new_xyz + ((size_t)b * S_ + s0) * 3;
    v2f a;
    if (lane < 16) { a[0] = q[lane * 3 + 0]; a[1] = q[lane * 3 + 1]; }
    else           { a[0] = q[(lane - 16) * 3 + 2]; a[1] = 1.0f; }

    const int col = lane & 15;
    const int rbase = (lane < 16) ? 0 : 8;

    for (int n0 = 0; n0 < N_; n0 += 32) {
        const float* x = xyz + ((size_t)b * N_ + n0) * 3;
        __builtin_prefetch(x + 32 * 3, 0, 0);

#pragma unroll
        for (int t = 0; t < 2; ++t) {
            const float* xt = x + t * 16 * 3;
            float px = xt[col * 3 + 0], py = xt[col * 3 + 1], pz = xt[col * 3 + 2];
            v2f bm;
            if (lane < 16) { bm[0] = -2.0f * px; bm[1] = -2.0f * py; }
            else           { bm[0] = -2.0f * pz; bm[1] = px * px + py * py + pz * pz; }

            v8f c = {};
            c = __builtin_amdgcn_wmma_f32_16x16x4_f32(
                    false, a, false, bm, (short)0, c, false, false);

            // Scatter C tile (M=r lanes 0-15, M=r+8 lanes 16-31) into LDS.
#pragma unroll
            for (int r = 0; r < 8; ++r)
                scw[(rbase + r) * 32 + t * 16 + col] = c[r];
        }
        asm volatile("s_wait_dscnt 0" ::: "memory"); // cross-lane LDS dep in-wave

        // All 32 lanes select: row = lane&15, columns half*16 .. half*16+15.
        const int cb = half * 16;
#pragma unroll
        for (int j = 0; j < 16; ++j) {
            float v = scw[row * 32 + cb + j];
            if (v < worst) {
                int vi = n0 + cb + j;
                int p = K_ - 1;
                while (p > 0) {
                    float pv = mylv[p - 1];
                    if (pv <= v) break;
                    mylv[p] = pv; myli[p] = myli[p - 1];
                    --p;
                }
                mylv[p] = v; myli[p] = vi;
                worst = mylv[K_ - 1];
            }
        }
        asm volatile("s_wait_dscnt 0" ::: "memory");
    }

    asm volatile("s_wait_dscnt 0" ::: "memory");
    // Merge the two sorted 32-sublists of each row; take 32 smallest.
    if (lane < 16) {
        float* v0 = lv + (row * 2 + 0) * K_; int* i0 = li + (row * 2 + 0) * K_;
        float* v1 = lv + (row * 2 + 1) * K_; int* i1 = li + (row * 2 + 1) * K_;
        int* o = knn_idx + ((size_t)(b * S_ + s0 + row)) * K_;
        int i = 0, j = 0;
        for (int k = 0; k < K_; ++k) {
            float va = v0[i], vb = v1[j];
            if (va <= vb) { o[k] = i0[i]; ++i; }
            else          { o[k] = i1[j]; ++j; }
        }
    }
}

// ---------------------------------------------------------------------------
// Kernel 3: per-(b,s) gather -> LDS, per-channel mean over K, accumulate
// sum of squared diffs into per-batch accumulator. (std over whole batch;
// diffs sum to zero over K per channel, so mean-of-diff = 0.)
// ---------------------------------------------------------------------------
__global__ void __launch_bounds__(64)
sumsq_kernel(const float* __restrict__ xyz, const float* __restrict__ points,
             const int* __restrict__ knn_idx, float* __restrict__ sumsq) {
    const int b = blockIdx.x / S_;
    const int s = blockIdx.x % S_;
    const int tid = threadIdx.x;

    __shared__ int sidx[K_];
    __shared__ float g[K_ * C67];
    __shared__ float cmean[C67];
    __shared__ float red[64];

    if (tid < K_) sidx[tid] = knn_idx[((size_t)(b * S_ + s)) * K_ + tid];
    __syncthreads();

    for (int e = tid; e < K_ * C67; e += 64) {
        int k = e / C67, c = e % C67;
        int id = sidx[k];
        g[e] = (c < D_) ? points[((size_t)b * N_ + id) * D_ + c]
                        : xyz[((size_t)b * N_ + id) * 3 + (c - D_)];
    }
    __syncthreads();
    for (int c = tid; c < C67; c += 64) {
        float sum = 0.0f;
#pragma unroll
        for (int k = 0; k < K_; ++k) sum += g[k * C67 + c];
        cmean[c] = sum * (1.0f / K_);
    }
    __syncthreads();

    float local = 0.0f;
    for (int e = tid; e < K_ * C67; e += 64) {
        float d = g[e] - cmean[e % C67];
        local += d * d;
    }
    red[tid] = local;
    __syncthreads();
#pragma unroll
    for (int off = 32; off > 0; off >>= 1) {
        if (tid < off) red[tid] += red[tid + off];
        __syncthreads();
    }
    if (tid == 0) atomicAdd(&sumsq[b], red[0]);
}

// ---------------------------------------------------------------------------
// Kernel 4: finalize — recompute gather+mean, normalize by batch std (ddof=1),
// apply affine, append anchor features; write output channels 0..130.
// ---------------------------------------------------------------------------
__global__ void __launch_bounds__(64)
finalize_kernel(const float* __restrict__ xyz, const float* __restrict__ points,
                const float* __restrict__ alpha, const float* __restrict__ beta,
                const int* __restrict__ knn_idx, const int* __restrict__ fps_idx,
                const float* __restrict__ sumsq, float* __restrict__ out_pts) {
    const int b = blockIdx.x / S_;
    const int s = blockIdx.x % S_;
    const int tid = threadIdx.x;

    __shared__ int sidx[K_];
    __shared__ float g[K_ * C67];
    __shared__ float cmean[C67];

    if (tid < K_) sidx[tid] = knn_idx[((size_t)(b * S_ + s)) * K_ + tid];
    __syncthreads();

    for (int e = tid; e < K_ * C67; e += 64) {
        int k = e / C67, c = e % C67;
        int id = sidx[k];
        g[e] = (c < D_) ? points[((size_t)b * N_ + id) * D_ + c]
                        : xyz[((size_t)b * N_ + id) * 3 + (c - D_)];
    }
    __syncthreads();
    for (int c = tid; c < C67; c += 64) {
        float sum = 0.0f;
#pragma unroll
        for (int k = 0; k < K_; ++k) sum += g[k * C67 + c];
        cmean[c] = sum * (1.0f / K_);
    }
    __syncthreads();

    const float cnt = (float)((size_t)S_ * K_ * C67 - 1);
    const float stdv = sqrtf(sumsq[b] / cnt);
    const float inv = 1.0f / (stdv + 1e-5f);

    float* o = out_pts + ((size_t)(b * S_ + s)) * K_ * COUT;
    for (int e = tid; e < K_ * C67; e += 64) {
        int k = e / C67, c = e % C67;
        o[(size_t)k * COUT + c] = alpha[c] * ((g[e] - cmean[c]) * inv) + beta[c];
    }
    const int aid = fps_idx[b * S_ + s];
    const float* anchor = points + ((size_t)b * N_ + aid) * D_;
    for (int e = tid; e < K_ * D_; e += 64) {
        int k = e / D_, c = e % D_;
        o[(size_t)k * COUT + C67 + c] = anchor[c];
    }
}

// ---------------------------------------------------------------------------
extern "C" void kernel_launch(void* const* d_in, const int* in_sizes, int n_in,
                              void* d_out, int out_size, void* d_ws, size_t ws_size,
                              hipStream_t stream) {
    const float* xyz    = (const float*)d_in[0]; // (B,N,3)
    const float* points = (const float*)d_in[1]; // (B,N,D)
    const float* alpha  = (const float*)d_in[2]; // (67,)
    const float* beta   = (const float*)d_in[3]; // (67,)

    float* out_newxyz = (float*)d_out;                       // B*S*3 floats
    float* out_pts    = out_newxyz + (size_t)B_ * S_ * 3;    // B*S*K*131 floats

    char* w = (char*)d_ws;
    int*   fps_idx = (int*)w;                                     // B*S ints
    int*   knn_idx = (int*)(w + (size_t)B_ * S_ * 4);             // B*S*K ints
    float* sumsq   = (float*)(w + (size_t)B_ * S_ * 4 + (size_t)B_ * S_ * K_ * 4);

    zero_acc_kernel<<<1, 32, 0, stream>>>(sumsq);
    fps_kernel<<<B_, 1024, 0, stream>>>(xyz, fps_idx, out_newxyz);
    knn_kernel<<<(B_ * S_ / 16) / 8, 256, 0, stream>>>(xyz, out_newxyz, knn_idx);
    sumsq_kernel<<<B_ * S_, 64, 0, stream>>>(xyz, points, knn_idx, sumsq);
    finalize_kernel<<<B_ * S_, 64, 0, stream>>>(xyz, points, alpha, beta,
                                                knn_idx, fps_idx, sumsq, out_pts);
}